// RelativePositionEmbedding_44375602102598
// MI455X (gfx1250) — compile-verified
//
#include <hip/hip_runtime.h>
#include <stdint.h>

// ---------------------------------------------------------------------------
// RelativePositionEmbedding: out[0,i,j,:] = (W.T)[clip(j-i,-32,32)+32,:] + b
// L = 2048, n_dims = 64, table = 65 x 64 f32 (16.6 KB)
// Roofline: 1.07 GB of f32 output @ 23.3 TB/s HBM => ~46 us store floor;
// ~0.03 FLOP/byte, so no matrix-core work exists here. CDNA5 paths used:
//   * TDM tensor_load_to_lds (async DMA of the W tile into LDS) + s_wait_tensorcnt
//   * bias-fused LDS table, ds_load_b128 reads (2-phase minimum for wave32)
//   * non-temporal global_store_b128 streaming stores (1 GB >> 192 MB L2)
// ---------------------------------------------------------------------------

typedef __attribute__((ext_vector_type(4))) float        v4f;
typedef __attribute__((ext_vector_type(4))) unsigned int u32x4;
typedef __attribute__((ext_vector_type(8))) int          i32x8;
typedef __attribute__((ext_vector_type(4))) int          i32x4;

#define RIDX_MAX 32
#define NTAB     65            // 2*RIDX_MAX + 1
#define ND       64            // n_dims
#define LSEQ     2048          // L
#define LOG2_L   11
#define V4_PER_PAIR (ND / 4)   // 16 float4 per (i,j)

// Grid-stride must be a multiple of 16*LSEQ (= 32768) float4s so that each
// thread's d4 and j are loop-invariant and i advances by a constant.
#define NBLOCKS  4096
#define NTHREADS 256

__global__ __launch_bounds__(NTHREADS)
void relpos_stream_kernel(const float* __restrict__ W,   // [ND][NTAB] row-major
                          const float* __restrict__ b,   // [ND]
                          float* __restrict__ out)       // [LSEQ][LSEQ][ND]
{
    __shared__ float stag[NTAB * ND];   // raw W staged by TDM (4160 f32)
    __shared__ float tbl [NTAB * ND];   // tbl[r*ND + d] = W[d][r] + b[d]

    // ---- Stage W into LDS via the Tensor Data Mover (async DMA) ----
#if __has_builtin(__builtin_amdgcn_tensor_load_to_lds)
    if (threadIdx.x < 32) {   // one wave issues the TDM op (TDM ignores EXEC)
        const uint64_t ga     = (uint64_t)(uintptr_t)W;
        const uint32_t ldsOff = (uint32_t)(uintptr_t)&stag[0]; // LDS address: low 32 bits of pointer

        // D# group 0: count=1 | lds_addr | global_addr[56:0] | type=2 ("image")
        u32x4 g0 = { 1u,
                     ldsOff,
                     (uint32_t)(ga & 0xFFFFFFFFu),
                     ((uint32_t)((ga >> 32) & 0x1FFFFFFu)) | (2u << 30) };

        // D# group 1: mask=0, data_size=2 (4B); tensor 65x64, tile 65x64, stride0=65
        i32x8 g1 = { (int)(2u << 16),                       // data_size = 4 bytes
                     (int)((uint32_t)NTAB << 16),           // tensor_dim0 = 65
                     (int)((uint32_t)ND   << 16),           // tensor_dim1 = 64
                     (int)((uint32_t)NTAB << 16),           // tile_dim0   = 65
                     (int)ND,                               // tile_dim1=64, tile_dim2=0
                     (int)NTAB,                             // tensor_dim0_stride = 65
                     (int)(((uint32_t)(NTAB * ND) & 0xFFFFu) << 16), // dim1_stride lo
                     0 };                                   // dim1_stride hi

        i32x4 gz = { 0, 0, 0, 0 };                          // groups 2/3 unused (2-D)
#if defined(__clang_major__) && (__clang_major__ >= 23)
        i32x8 gz8 = { 0, 0, 0, 0, 0, 0, 0, 0 };
        __builtin_amdgcn_tensor_load_to_lds(g0, g1, gz, gz, gz8, 0);
#else
        __builtin_amdgcn_tensor_load_to_lds(g0, g1, gz, gz, 0);
#endif
        __builtin_amdgcn_s_wait_tensorcnt(0);
    }
#else
    for (int t = threadIdx.x; t < NTAB * ND; t += NTHREADS) stag[t] = W[t];
#endif
    __syncthreads();

    // ---- Build bias-fused, transposed table: tbl[r][d] = W[d][r] + b[d] ----
    for (int t = threadIdx.x; t < NTAB * ND; t += NTHREADS) {
        const int r = t >> 6;          // t / 64
        const int d = t & (ND - 1);    // t % 64
        tbl[(r << 6) + d] = stag[d * NTAB + r] + b[d];
    }
    __syncthreads();

    // ---- Streaming phase ----
    // v indexes float4s of the output. With stride = 2^20, per-thread d4 and j
    // are invariant; i advances by stride>>15 (= 32) with no carries into j/d4.
    const v4f* __restrict__ tbl4 = (const v4f*)tbl;
    v4f* __restrict__ out4 = (v4f*)out;

    const size_t totalVec = (size_t)LSEQ * (size_t)LSEQ * (size_t)V4_PER_PAIR; // 64 Mi
    const size_t stride   = (size_t)NBLOCKS * NTHREADS;                        // 2^20
    const size_t v0       = (size_t)blockIdx.x * blockDim.x + threadIdx.x;

    const unsigned d4 = (unsigned)v0 & (V4_PER_PAIR - 1);    // dim slot, invariant
    const unsigned j  = (unsigned)(v0 >> 4) & (LSEQ - 1);    // column, invariant
    unsigned       i  = (unsigned)((v0 >> 4) >> LOG2_L);     // row, += 32 per iter
    const unsigned di = (unsigned)(stride >> (4 + LOG2_L));  // = 32

    const v4f* __restrict__ rowBase = tbl4 + d4;

    for (size_t v = v0; v < totalVec; v += stride) {
        int r = (int)j - (int)i;                              // clamp -> v_med3
        r = (r < -RIDX_MAX) ? -RIDX_MAX : ((r > RIDX_MAX) ? RIDX_MAX : r);
        const v4f val = rowBase[(unsigned)(r + RIDX_MAX) << 4]; // ds_load_b128
        __builtin_nontemporal_store(val, out4 + v);           // global_store_b128 NT
        i += di;
    }
}

// ---------------------------------------------------------------------------
// Harness entry. d_in order: [0]=aa_seq (int32, unused by the math),
// [1]=W (f32, 64*65), [2]=b (f32, 64). d_out: f32, 2048*2048*64.
// ---------------------------------------------------------------------------
extern "C" void kernel_launch(void* const* d_in, const int* in_sizes, int n_in,
                              void* d_out, int out_size, void* d_ws, size_t ws_size,
                              hipStream_t stream)
{
    (void)in_sizes; (void)n_in; (void)out_size; (void)d_ws; (void)ws_size;
    const float* W = (const float*)d_in[1];
    const float* b = (const float*)d_in[2];
    float*     out = (float*)d_out;

    dim3 grid(NBLOCKS), block(NTHREADS);   // 1 Mi threads, 64 float4-stores each
    hipLaunchKernelGGL(relpos_stream_kernel, grid, block, 0, stream, W, b, out);
}